// SphericalExpansion_31731218382983
// MI455X (gfx1250) — compile-verified
//
#include <hip/hip_runtime.h>
#include <math.h>

// Spherical expansion + segmented reduction via one-hot fp32 WMMA (gfx1250).
//
// out[l][atom, m, n, c] = sum_{p: i[p]=atom} ang[p, offm_l+m] * rad[p, offn_l+n] * sp[p, c]
//
// Segment-sum as GEMM over sorted edges: for each 16-atom subtile of a block's
// atom span,  C(16 atoms x 16 feats) += S(16x4 one-hot) x E(4x16)  with
// V_WMMA_F32_16X16X4_F32, K = EPB edges in steps of 4.
//
// Phase 1 pre-fuses angsp[edge][m*4+c] = ang[m]*sp[c] so each B element is
// rad * angsp (2 DS reads + 1 mul). One-hot A registers are built once per
// atom subtile and reused across all feature tiles handled by the wave.

#define EPB 112           // edges per block (112*132*4 B + seg = ~59.6 KB LDS)
#define KG  (EPB / 4)     // K-groups of 4 edges
#define NFT 63            // 1008 features / 16 per N-tile
#define NWAVES 8          // 256 threads / wave32

typedef __attribute__((ext_vector_type(2))) float v2f;
typedef __attribute__((ext_vector_type(8))) float v8f;

__global__ void zero_f32(float* __restrict__ o, int n) {
  int idx = blockIdx.x * blockDim.x + threadIdx.x;
  if (idx < n) o[idx] = 0.0f;
}

__device__ __forceinline__ float freq_of(int q) {
  // freqs = concat(1..20, 1..18, 1..16, 1..14)
  int f;
  if (q < 20)      f = q + 1;
  else if (q < 38) f = q - 19;
  else if (q < 54) f = q - 37;
  else             f = q - 53;
  return (float)f;
}

// Map flat feature column g in [0,1008) -> indices.
// l0: ml=1,nl=20 (cols 0..79)   l1: ml=3,nl=18 (80..295)
// l2: ml=5,nl=16 (296..615)     l3: ml=7,nl=14 (616..1007)
__device__ __forceinline__ void decomp(int g, int b1, int b2, int b3,
                                       int& loc, int& radIdx, int& angIdx,
                                       int& ch, int& outBase, int& astride) {
  int base, nl4, offn, offm;
  if (g < 80)       { base = 0;   nl4 = 80; offn = 0;  offm = 0; outBase = 0;  astride = 80;  }
  else if (g < 296) { base = 80;  nl4 = 72; offn = 20; offm = 1; outBase = b1; astride = 216; }
  else if (g < 616) { base = 296; nl4 = 64; offn = 38; offm = 4; outBase = b2; astride = 320; }
  else              { base = 616; nl4 = 56; offn = 54; offm = 9; outBase = b3; astride = 392; }
  loc = g - base;
  int m   = loc / nl4;
  int rem = loc - m * nl4;
  int n   = rem >> 2;
  ch      = rem & 3;
  radIdx  = offn + n;
  angIdx  = offm + m;
}

__global__ __launch_bounds__(256) void sph_expand_wmma(
    const float* __restrict__ Rij, const int* __restrict__ segi,
    const int* __restrict__ nbrj, const int* __restrict__ species,
    const float* __restrict__ emb, float* __restrict__ out,
    int n_edges, int n_atoms, int b1, int b2, int b3)
{
  __shared__ float ls_rad[EPB][68];   // radial basis per edge
  __shared__ float ls_asp[EPB][64];   // ang[m]*sp[c], flat m*4+c
  __shared__ int   ls_seg[EPB];       // destination atom per edge

  const int t  = threadIdx.x;
  const int e0 = blockIdx.x * EPB;

  // ---------------- Phase 1: per-edge basis into LDS (2 threads / edge) ----
  if (t < 2 * EPB) {
    const int le   = t >> 1;
    const int half = t & 1;
    const int e    = e0 + le;
    if (e < n_edges) {
      float x = Rij[3 * e + 0], y = Rij[3 * e + 1], z = Rij[3 * e + 2];
      float r  = sqrtf(x * x + y * y + z * z);
      const float PI5 = 0.6283185307179586f;      // pi / CUTOFF
      if (half == 0) {
        float rc = fminf(r, 5.0f);
        float fc = 0.5f * (__cosf(rc * PI5) + 1.0f);
        float w  = r * PI5;
        for (int q = 0; q < 68; ++q) ls_rad[le][q] = __sinf(freq_of(q) * w) * fc;
      } else {
        float inv = 1.0f / (r + 1e-12f);
        float ux = x * inv, uy = y * inv, uz = z * inv;
        float x2 = ux * ux, y2 = uy * uy, z2 = uz * uz;
        float ang[16];
        ang[0]  = 0.28209479177387814f;
        ang[1]  = 0.4886025119029199f * uy;
        ang[2]  = 0.4886025119029199f * uz;
        ang[3]  = 0.4886025119029199f * ux;
        ang[4]  = 1.0925484305920792f * ux * uy;
        ang[5]  = 1.0925484305920792f * uy * uz;
        ang[6]  = 0.31539156525252005f * (3.0f * z2 - 1.0f);
        ang[7]  = 1.0925484305920792f * ux * uz;
        ang[8]  = 0.5462742152960396f * (x2 - y2);
        ang[9]  = 0.5900435899266435f * uy * (3.0f * x2 - y2);
        ang[10] = 2.890611442640554f * ux * uy * uz;
        ang[11] = 0.4570457994644658f * uy * (5.0f * z2 - 1.0f);
        ang[12] = 0.3731763325901154f * uz * (5.0f * z2 - 3.0f);
        ang[13] = 0.4570457994644658f * ux * (5.0f * z2 - 1.0f);
        ang[14] = 1.445305721320277f * uz * (x2 - y2);
        ang[15] = 0.5900435899266435f * ux * (x2 - 3.0f * y2);
        const int sj = species[nbrj[e]];
        float sp[4];
        sp[0] = emb[4 * sj + 0]; sp[1] = emb[4 * sj + 1];
        sp[2] = emb[4 * sj + 2]; sp[3] = emb[4 * sj + 3];
        for (int m = 0; m < 16; ++m)
          for (int c = 0; c < 4; ++c)
            ls_asp[le][m * 4 + c] = ang[m] * sp[c];
        ls_seg[le] = segi[e];
      }
    } else {
      if (half == 0) {
        for (int q = 0; q < 68; ++q) ls_rad[le][q] = 0.0f;
      } else {
        for (int q = 0; q < 64; ++q) ls_asp[le][q] = 0.0f;
        ls_seg[le] = -1;                // never matches a one-hot row
      }
    }
  }
  __syncthreads();

  // ---------------- Phase 2: segmented GEMM via WMMA -----------------------
  const int evalid = min(EPB, n_edges - e0);
  const int a0 = ls_seg[0];
  const int a1 = ls_seg[evalid - 1];
  const int nsub = (a1 - a0 + 16) >> 4;      // ceil(span/16)

  const int wave = t >> 5;
  const int lane = t & 31;
  const int mrow = lane & 15;                // A: M row; B/C: N column
  const int khi  = (lane & 16) ? 2 : 0;      // K offset for upper half-wave
  const int radd = (lane & 16) ? 8 : 0;      // C: row offset for upper half

  for (int sub = 0; sub < nsub; ++sub) {
    const int rowBase = a0 + (sub << 4);
    const int myAtomRow = rowBase + mrow;

    // One-hot A registers for the whole K range, reused across feature tiles.
    v2f areg[KG];
#pragma unroll
    for (int i = 0; i < KG; ++i) {
      const int k0 = (i << 2) + khi;
      areg[i].x = (ls_seg[k0]     == myAtomRow) ? 1.0f : 0.0f;
      areg[i].y = (ls_seg[k0 + 1] == myAtomRow) ? 1.0f : 0.0f;
    }

    for (int ft = wave; ft < NFT; ft += NWAVES) {
      const int g = (ft << 4) + mrow;
      int loc, radIdx, angIdx, ch, outBase, astride;
      decomp(g, b1, b2, b3, loc, radIdx, angIdx, ch, outBase, astride);
      const int aspIdx = (angIdx << 2) | ch;

      v8f acc = {0.f, 0.f, 0.f, 0.f, 0.f, 0.f, 0.f, 0.f};
#pragma unroll
      for (int i = 0; i < KG; ++i) {
        const int k0 = (i << 2) + khi;
        v2f bv;
        bv.x = ls_rad[k0][radIdx]     * ls_asp[k0][aspIdx];
        bv.y = ls_rad[k0 + 1][radIdx] * ls_asp[k0 + 1][aspIdx];
        acc = __builtin_amdgcn_wmma_f32_16x16x4_f32(
            /*neg_a=*/false, areg[i], /*neg_b=*/false, bv,
            /*c_mod=*/(short)0, acc, /*reuse_a=*/false, /*reuse_b=*/false);
      }

      // Flush C tile: VGPR r holds M=r (lanes 0-15) and M=r+8 (lanes 16-31)
#pragma unroll
      for (int rr = 0; rr < 8; ++rr) {
        const int atom = rowBase + rr + radd;
        if (atom <= a1 && atom < n_atoms) {
          atomicAdd(&out[outBase + atom * astride + loc], acc[rr]);
        }
      }
    }
  }
}

extern "C" void kernel_launch(void* const* d_in, const int* in_sizes, int n_in,
                              void* d_out, int out_size, void* d_ws, size_t ws_size,
                              hipStream_t stream) {
  const float* Rij     = (const float*)d_in[0];
  const int*   segi    = (const int*)d_in[1];
  const int*   nbrj    = (const int*)d_in[2];
  const int*   species = (const int*)d_in[3];
  const float* emb     = (const float*)d_in[4];
  float*       out     = (float*)d_out;

  const int n_edges = in_sizes[1];
  const int n_atoms = in_sizes[3];
  const int b1 = n_atoms * 80;
  const int b2 = b1 + n_atoms * 216;
  const int b3 = b2 + n_atoms * 320;

  zero_f32<<<(out_size + 255) / 256, 256, 0, stream>>>(out, out_size);

  const int nblk = (n_edges + EPB - 1) / EPB;
  sph_expand_wmma<<<nblk, 256, 0, stream>>>(Rij, segi, nbrj, species, emb, out,
                                            n_edges, n_atoms, b1, b2, b3);
}